// GCNNet_22625887715477
// MI455X (gfx1250) — compile-verified
//
#include <hip/hip_runtime.h>
#include <hip/hip_bf16.h>

typedef float v2f __attribute__((ext_vector_type(2)));
typedef float v8f __attribute__((ext_vector_type(8)));

#define HID 128

// ---------------------------------------------------------------------------
// Degree / normalization kernels
// ---------------------------------------------------------------------------
__global__ __launch_bounds__(256) void k_init_deg(float* __restrict__ deg, int n) {
    int i = blockIdx.x * blockDim.x + threadIdx.x;
    if (i < n) deg[i] = 1.0f;  // self loop contributes 1
}

__global__ __launch_bounds__(256) void k_count_deg(const int* __restrict__ col,
                                                   float* __restrict__ deg, int E) {
    int e = blockIdx.x * blockDim.x + threadIdx.x;
    if (e < E) atomicAdd(&deg[col[e]], 1.0f);
}

__global__ __launch_bounds__(256) void k_make_dis(float* __restrict__ deg, int n) {
    int i = blockIdx.x * blockDim.x + threadIdx.x;
    if (i < n) deg[i] = rsqrtf(deg[i]);  // deg >= 1 always (self loops)
}

// ---------------------------------------------------------------------------
// Small weight transpose: Wt[OUT][IN] = W[IN][OUT]  (makes B fragments
// contiguous in K so the GEMM inner loop does one b64 load per operand)
// ---------------------------------------------------------------------------
template <int IN, int OUT>
__global__ __launch_bounds__(256) void k_transpose(const float* __restrict__ W,
                                                   float* __restrict__ Wt) {
    int i = blockIdx.x * blockDim.x + threadIdx.x;
    if (i < IN * OUT) {
        int k = i / OUT, n = i % OUT;
        Wt[n * IN + k] = W[i];
    }
}

// ---------------------------------------------------------------------------
// Fast WMMA fp32 GEMM for full tiles: out[N x OUT] = act(H) @ W, W given
// transposed as Wt[OUT x IN]. Requires IN % 4 == 0 and OUT % 16 == 0.
// One wave = one 16x16 output tile; V_WMMA_F32_16X16X4_F32, K step 4.
// Steady state inner loop: 1x b64 (A) + 1x b64 (B) + 1x wmma.
//
// A fragment (16x4 fp32): lanes 0-15 -> M=lane, K={k,k+1};
//                         lanes 16-31 -> M=lane-16, K={k+2,k+3}.
// B fragment (4x16) mirrors it (half selects the K pair).
// ---------------------------------------------------------------------------
template <int IN, int OUT, int RELU_IN>
__global__ __launch_bounds__(32) void k_gemm_wmma_fast(const float* __restrict__ H,
                                                       const float* __restrict__ Wt,
                                                       float* __restrict__ out) {
    const int lane = threadIdx.x & 31;
    const int half = lane >> 4;
    const int l    = lane & 15;
    const int m0   = blockIdx.x << 4;   // N is a multiple of 16
    const int n0   = blockIdx.y << 4;

    const float* __restrict__ Arow = H  + (size_t)(m0 + l) * IN;
    const float* __restrict__ Brow = Wt + (size_t)(n0 + l) * IN;

    v8f c = {};
#pragma unroll
    for (int k = 0; k < IN; k += 4) {
        const int kk = k + 2 * half;
        v2f a = *(const v2f*)(Arow + kk);        // 8B aligned: IN even, kk even
        if (RELU_IN) { a.x = fmaxf(a.x, 0.0f); a.y = fmaxf(a.y, 0.0f); }
        v2f b = *(const v2f*)(Brow + kk);
        c = __builtin_amdgcn_wmma_f32_16x16x4_f32(
                false, a, false, b, (short)0, c, false, false);
    }

    // C/D layout: VGPR r -> M = r (lanes 0-15) / r+8 (lanes 16-31), N = lane&15.
#pragma unroll
    for (int r = 0; r < 8; ++r)
        out[(size_t)(m0 + r + half * 8) * OUT + n0 + l] = c[r];
}

// ---------------------------------------------------------------------------
// Generic (masked) WMMA GEMM for ragged shapes (IN=3 layer, OUT=1 layer).
// Guards are clamped addresses + selects -> EXEC stays all-ones for WMMA.
// ---------------------------------------------------------------------------
template <int IN, int OUT, int RELU_IN>
__global__ __launch_bounds__(32) void k_gemm_wmma_gen(const float* __restrict__ H,
                                                      const float* __restrict__ W,
                                                      float* __restrict__ out) {
    const int lane = threadIdx.x & 31;
    const int half = lane >> 4;
    const int l    = lane & 15;
    const int m0   = blockIdx.x << 4;
    const int n0   = blockIdx.y << 4;

    const int   row     = m0 + l;
    const int   col     = n0 + l;
    const int   colc    = (col < OUT) ? col : 0;
    const float colmask = (col < OUT) ? 1.0f : 0.0f;

    v8f c = {};
#pragma unroll
    for (int k = 0; k < IN; k += 4) {
        const int kk  = k + 2 * half;
        const int k0  = (kk     < IN) ? kk       : (IN - 1);
        const int k1  = (kk + 1 < IN) ? (kk + 1) : (IN - 1);
        const float km0 = (kk     < IN) ? 1.0f : 0.0f;
        const float km1 = (kk + 1 < IN) ? 1.0f : 0.0f;

        float a0 = H[row * IN + k0];
        float a1 = H[row * IN + k1];
        if (RELU_IN) { a0 = fmaxf(a0, 0.0f); a1 = fmaxf(a1, 0.0f); }
        v2f a = { a0 * km0, a1 * km1 };

        float b0 = W[k0 * OUT + colc];
        float b1 = W[k1 * OUT + colc];
        v2f b = { b0 * colmask * km0, b1 * colmask * km1 };

        c = __builtin_amdgcn_wmma_f32_16x16x4_f32(
                false, a, false, b, (short)0, c, false, false);
    }

#pragma unroll
    for (int r = 0; r < 8; ++r) {
        const int mrow = m0 + r + half * 8;
        if (col < OUT) out[(size_t)mrow * OUT + col] = c[r];
    }
}

// ---------------------------------------------------------------------------
// out[i,f] = dis[i]^2 * tmp[i,f] + b[f]   (self-loop term + bias; also
// initializes the destination so the edge scatter can accumulate). float4.
// ---------------------------------------------------------------------------
template <int OUT>
__global__ __launch_bounds__(256) void k_self_bias4(const float4* __restrict__ tmp,
                                                    const float* __restrict__ dis,
                                                    const float* __restrict__ b,
                                                    float4* __restrict__ out, int n) {
    int i = blockIdx.x * blockDim.x + threadIdx.x;      // over n * OUT/4
    int node = i / (OUT / 4);
    int f4   = (i % (OUT / 4)) * 4;
    if (node < n) {
        float d  = dis[node];
        float d2 = d * d;
        float4 t = tmp[i];
        float4 o;
        o.x = d2 * t.x + b[f4 + 0];
        o.y = d2 * t.y + b[f4 + 1];
        o.z = d2 * t.z + b[f4 + 2];
        o.w = d2 * t.w + b[f4 + 3];
        out[i] = o;
    }
}

__global__ __launch_bounds__(256) void k_self_bias1(const float* __restrict__ tmp,
                                                    const float* __restrict__ dis,
                                                    const float* __restrict__ b,
                                                    float* __restrict__ out, int n) {
    int i = blockIdx.x * blockDim.x + threadIdx.x;
    if (i < n) {
        float d = dis[i];
        out[i] = d * d * tmp[i] + b[0];
    }
}

// ---------------------------------------------------------------------------
// Edge scatter: out[col, f..f+3] += dis[row]*dis[col] * tmp[row, f..f+3]
// float4 gather (L2-resident; tmp = 51MB < 192MB L2), 4 f32 atomics.
// ---------------------------------------------------------------------------
template <int OUT>
__global__ __launch_bounds__(256) void k_scatter4(const int* __restrict__ rowi,
                                                  const int* __restrict__ coli,
                                                  const float* __restrict__ dis,
                                                  const float4* __restrict__ tmp,
                                                  float* __restrict__ out, int E) {
    const int Q = OUT / 4;
    int t = blockIdx.x * blockDim.x + threadIdx.x;      // over E * Q
    int e = t / Q;
    int q = t % Q;
    if (e < E) {
        int r = rowi[e];
        int c = coli[e];
        float nrm = dis[r] * dis[c];
        float4 v = tmp[(size_t)r * Q + q];
        float* dst = out + (size_t)c * OUT + q * 4;
        atomicAdd(dst + 0, nrm * v.x);
        atomicAdd(dst + 1, nrm * v.y);
        atomicAdd(dst + 2, nrm * v.z);
        atomicAdd(dst + 3, nrm * v.w);
    }
}

__global__ __launch_bounds__(256) void k_scatter1(const int* __restrict__ rowi,
                                                  const int* __restrict__ coli,
                                                  const float* __restrict__ dis,
                                                  const float* __restrict__ tmp,
                                                  float* __restrict__ out, int E) {
    int e = blockIdx.x * blockDim.x + threadIdx.x;
    if (e < E) {
        int r = rowi[e];
        int c = coli[e];
        atomicAdd(&out[c], dis[r] * dis[c] * tmp[r]);
    }
}

// ---------------------------------------------------------------------------
// Host-side launcher
// ---------------------------------------------------------------------------
extern "C" void kernel_launch(void* const* d_in, const int* in_sizes, int n_in,
                              void* d_out, int out_size, void* d_ws, size_t ws_size,
                              hipStream_t stream) {
    const float* x  = (const float*)d_in[0];
    const int*   ei = (const int*)d_in[1];
    const float* W1 = (const float*)d_in[2];
    const float* b1 = (const float*)d_in[3];
    const float* W2 = (const float*)d_in[4];
    const float* b2 = (const float*)d_in[5];
    const float* W3 = (const float*)d_in[6];
    const float* b3 = (const float*)d_in[7];
    const float* W4 = (const float*)d_in[8];
    const float* b4 = (const float*)d_in[9];
    const float* W5 = (const float*)d_in[10];
    const float* b5 = (const float*)d_in[11];
    float* outp = (float*)d_out;

    const int N = in_sizes[0] / 3;   // 100000 (multiple of 16)
    const int E = in_sizes[1] / 2;   // 600000
    const int* rowi = ei;            // edge_index[0]: sources
    const int* coli = ei + E;        // edge_index[1]: targets

    // Workspace (floats): dis[N] | tmp[N*128] | hA[N*128] | hB[N*128] |
    //                     Wt2[128*128] | Wt3[64*128] | Wt4[64*64]
    float* dis = (float*)d_ws;
    float* tmp = dis + N;
    float* hA  = tmp + (size_t)N * HID;
    float* hB  = hA  + (size_t)N * HID;
    float* Wt2 = hB  + (size_t)N * HID;
    float* Wt3 = Wt2 + 128 * 128;
    float* Wt4 = Wt3 + 128 * 64;

    const int MT = (N + 15) / 16;   // row tiles

    // --- degree normalization + weight transposes ----------------------------
    k_init_deg<<<(N + 255) / 256, 256, 0, stream>>>(dis, N);
    k_count_deg<<<(E + 255) / 256, 256, 0, stream>>>(coli, dis, E);
    k_make_dis<<<(N + 255) / 256, 256, 0, stream>>>(dis, N);
    k_transpose<128, 128><<<(128 * 128 + 255) / 256, 256, 0, stream>>>(W2, Wt2);
    k_transpose<128, 64><<<(128 * 64 + 255) / 256, 256, 0, stream>>>(W3, Wt3);
    k_transpose<64, 64><<<(64 * 64 + 255) / 256, 256, 0, stream>>>(W4, Wt4);

    // --- layer 1: x[N,3] @ W1 -> hA[N,128] (generic: IN=3) -------------------
    k_gemm_wmma_gen<3, 128, 0><<<dim3(MT, 8), 32, 0, stream>>>(x, W1, tmp);
    k_self_bias4<128><<<((size_t)N * 32 + 255) / 256, 256, 0, stream>>>(
        (const float4*)tmp, dis, b1, (float4*)hA, N);
    k_scatter4<128><<<((size_t)E * 32 + 255) / 256, 256, 0, stream>>>(
        rowi, coli, dis, (const float4*)tmp, hA, E);

    // --- layer 2: relu(hA)[N,128] @ W2 -> hB[N,128] --------------------------
    k_gemm_wmma_fast<128, 128, 1><<<dim3(MT, 8), 32, 0, stream>>>(hA, Wt2, tmp);
    k_self_bias4<128><<<((size_t)N * 32 + 255) / 256, 256, 0, stream>>>(
        (const float4*)tmp, dis, b2, (float4*)hB, N);
    k_scatter4<128><<<((size_t)E * 32 + 255) / 256, 256, 0, stream>>>(
        rowi, coli, dis, (const float4*)tmp, hB, E);

    // --- layer 3: relu(hB)[N,128] @ W3 -> hA[N,64] ---------------------------
    k_gemm_wmma_fast<128, 64, 1><<<dim3(MT, 4), 32, 0, stream>>>(hB, Wt3, tmp);
    k_self_bias4<64><<<((size_t)N * 16 + 255) / 256, 256, 0, stream>>>(
        (const float4*)tmp, dis, b3, (float4*)hA, N);
    k_scatter4<64><<<((size_t)E * 16 + 255) / 256, 256, 0, stream>>>(
        rowi, coli, dis, (const float4*)tmp, hA, E);

    // --- layer 4: relu(hA)[N,64] @ W4 -> hB[N,64] ----------------------------
    k_gemm_wmma_fast<64, 64, 1><<<dim3(MT, 4), 32, 0, stream>>>(hA, Wt4, tmp);
    k_self_bias4<64><<<((size_t)N * 16 + 255) / 256, 256, 0, stream>>>(
        (const float4*)tmp, dis, b4, (float4*)hB, N);
    k_scatter4<64><<<((size_t)E * 16 + 255) / 256, 256, 0, stream>>>(
        rowi, coli, dis, (const float4*)tmp, hB, E);

    // --- layer 5: relu(hB)[N,64] @ W5 -> out[N,1] (no final relu) ------------
    k_gemm_wmma_gen<64, 1, 1><<<dim3(MT, 1), 32, 0, stream>>>(hB, W5, tmp);
    k_self_bias1<<<(N + 255) / 256, 256, 0, stream>>>(tmp, dis, b5, outp, N);
    k_scatter1<<<(E + 255) / 256, 256, 0, stream>>>(rowi, coli, dis, tmp, outp, E);
}